// QuantumLayer_79431125172735
// MI455X (gfx1250) — compile-verified
//
#include <hip/hip_runtime.h>
#include <hip/hip_bf16.h>

// ---------------------------------------------------------------------------
// 8-qubit batched circuit on MI455X (gfx1250, wave32).
//
//   out[b,q] = sum_j sign_q(j) * | (U(theta) @ enc(x_b))_j |^2
//
// enc(x_b) is a REAL product state (RY encoding of |0..0>); U is the
// batch-invariant 256x256 complex variational unitary.
//
// Kernel A: build U (transposed, row j contiguous over k) into d_ws.
// Kernel B: [16384,256] x [256,256] real-x-complex GEMM on the full-f32
//           matrix pipe (V_WMMA_F32_16X16X4_F32), with U streamed through
//           LDS via double-buffered GLOBAL_LOAD_ASYNC_TO_LDS_B128
//           (ASYNCcnt / s_wait_asynccnt), then |.|^2 and the signed
//           Z-expectation reduction.
// ---------------------------------------------------------------------------

#define N_QUBITS 8
#define DIM      256        // 2^8 amplitudes
#define N_LAYERS 2
#define BATCH    16384

#define KC       16                 // K-chunk staged per async round
#define NCHUNK   (DIM / KC)         // 16 chunks
#define KS       20                 // padded LDS row stride (floats):
                                    //  - multiple of 4 -> 16B-aligned b128 stores
                                    //  - 20*m mod 64 distinct for m=0..15 -> no bank conflicts

typedef __attribute__((ext_vector_type(2))) float v2f;   // f32 WMMA A/B frag (2 VGPRs)
typedef __attribute__((ext_vector_type(8))) float v8f;   // f32 WMMA C/D frag (8 VGPRs)
typedef __attribute__((ext_vector_type(4))) int   v4i;   // 16B async-copy payload

#if defined(__has_builtin)
#if __has_builtin(__builtin_amdgcn_global_load_async_to_lds_b128) && \
    __has_builtin(__builtin_amdgcn_s_wait_asynccnt)
#define QC_ASYNC_LDS 1
#endif
#endif
#ifndef QC_ASYNC_LDS
#define QC_ASYNC_LDS 0
#endif

#if QC_ASYNC_LDS
typedef __attribute__((address_space(1))) v4i gv4i;      // global 16B payload
typedef __attribute__((address_space(3))) v4i lv4i;      // LDS    16B payload
#endif

// Wire q maps to bit (7-q) of the flattened amplitude index (wire 0 = MSB,
// matching the reference's reshape (B,2,2,...,2) with wire q -> axis q+1).

// ========================= Kernel A: build U(theta) ========================
// One block per basis column k: simulate the variational circuit on e_k in
// LDS (256 complex amplitudes, 128 pairs / 128 threads per 1q gate).
__global__ __launch_bounds__(128)
void qc_build_unitary(const float* __restrict__ theta,   // [2,8,3]
                      float* __restrict__ UrT,           // [256 j][256 k]
                      float* __restrict__ UiT) {         // [256 j][256 k]
  __shared__ float sr[DIM];
  __shared__ float si[DIM];
  const int k = blockIdx.x;
  const int t = threadIdx.x;

  for (int i = t; i < DIM; i += 128) { sr[i] = (i == k) ? 1.f : 0.f; si[i] = 0.f; }

  for (int layer = 0; layer < N_LAYERS; ++layer) {
    // ---- Rot(phi,th,om) = RZ(om) RY(th) RZ(phi) on each wire ----
    for (int q = 0; q < N_QUBITS; ++q) {
      const float phi = theta[(layer * 8 + q) * 3 + 0];
      const float th  = theta[(layer * 8 + q) * 3 + 1];
      const float om  = theta[(layer * 8 + q) * 3 + 2];
      float ct, st;  sincosf(0.5f * th, &st, &ct);
      float ca, sa;  sincosf(0.5f * (phi + om), &sa, &ca);
      float cb, sb;  sincosf(0.5f * (phi - om), &sb, &cb);
      // u00 = e^{-i(phi+om)/2} ct     u01 = -e^{+i(phi-om)/2} st
      // u10 = e^{-i(phi-om)/2} st     u11 = e^{+i(phi+om)/2} ct
      const float u00r =  ca * ct, u00i = -sa * ct;
      const float u01r = -cb * st, u01i = -sb * st;
      const float u10r =  cb * st, u10i = -sb * st;
      const float u11r =  ca * ct, u11i =  sa * ct;

      __syncthreads();                      // previous gate's writes visible
      const int p    = 7 - q;               // paired bit position
      const int mask = (1 << p) - 1;
      const int i0   = ((t >> p) << (p + 1)) | (t & mask);
      const int i1   = i0 | (1 << p);
      const float a0r = sr[i0], a0i = si[i0];
      const float a1r = sr[i1], a1i = si[i1];
      sr[i0] = u00r * a0r - u00i * a0i + u01r * a1r - u01i * a1i;
      si[i0] = u00r * a0i + u00i * a0r + u01r * a1i + u01i * a1r;
      sr[i1] = u10r * a0r - u10i * a0i + u11r * a1r - u11i * a1i;
      si[i1] = u10r * a0i + u10i * a0r + u11r * a1i + u11i * a1r;
    }
    // ---- CNOT chain: control q (bit 7-q) -> target q+1 (bit 6-q) ----
    for (int q = 0; q < N_QUBITS - 1; ++q) {
      __syncthreads();
      if (t < 64) {
        const int pt  = 6 - q;              // target bit
        const int pc  = pt + 1;             // control bit (adjacent, higher)
        const int low = t & ((1 << pt) - 1);
        const int i0  = ((t >> pt) << (pt + 2)) | (1 << pc) | low;  // c=1,t=0
        const int i1  = i0 | (1 << pt);                             // c=1,t=1
        const float tr = sr[i0], ti = si[i0];
        sr[i0] = sr[i1]; si[i0] = si[i1];
        sr[i1] = tr;     si[i1] = ti;
      }
    }
  }

  __syncthreads();
  for (int j = t; j < DIM; j += 128) {
    UrT[(size_t)j * DIM + k] = sr[j];
    UiT[(size_t)j * DIM + k] = si[j];
  }
}

// ===================== Kernel B: batched apply + <Z_q> =====================
// Block = 16 samples (M-tile), 256 threads = 8 waves. Each wave owns two
// 16-wide j-tiles and accumulates Re/Im via V_WMMA_F32_16X16X4_F32, with U
// double-buffer staged into LDS by async copies (ASYNCcnt pipeline).
__global__ __launch_bounds__(256)
void qc_batch_apply(const float* __restrict__ x,     // [BATCH,8]
                    const float* __restrict__ UrT,   // [256 j][256 k]
                    const float* __restrict__ UiT,
                    float* __restrict__ out) {       // [BATCH,8]
  __shared__ float A_lds[16][DIM];        // encoded real states (M x K)   16 KB
  __shared__ float P_lds[16][DIM];        // probabilities (M x N)         16 KB
  __shared__ float UrC[2][DIM][KS];       // staged Re(U) chunk, 2 bufs    40 KB
  __shared__ float UiC[2][DIM][KS];       // staged Im(U) chunk, 2 bufs    40 KB
  __shared__ float c_l[16][8];
  __shared__ float s_l[16][8];

  const int t  = threadIdx.x;
  const int b0 = blockIdx.x * 16;

  // ---- Phase 1a: per-(sample, wire) sincos of 0.5*pi*clip(x) ----
  if (t < 128) {
    const int sm = t >> 3, q = t & 7;
    float xv = x[(size_t)(b0 + sm) * 8 + q];
    xv = fminf(fmaxf(xv, -1.f), 1.f);
    float s, c;  sincosf(1.57079632679489662f * xv, &s, &c);
    c_l[sm][q] = c;  s_l[sm][q] = s;
  }
  __syncthreads();

  // ---- Phase 1b: encoded product-state amplitudes A[m][j] ----
  {
    const int sm = t >> 4;
    const int jb = (t & 15) << 4;
    for (int jj = 0; jj < 16; ++jj) {
      const int j = jb + jj;
      float a = 1.f;
#pragma unroll
      for (int q = 0; q < N_QUBITS; ++q)
        a *= ((j >> (7 - q)) & 1) ? s_l[sm][q] : c_l[sm][q];
      A_lds[sm][j] = a;
    }
  }

  // ---- Async stage of U chunk k0 -> buffer buf (cooperative, 256 thr) ----
  // Per matrix per chunk: 256 rows x 16 floats = 1024 x b128 transfers.
  auto stage = [&](int buf, int k0) {
#pragma unroll
    for (int i = 0; i < 4; ++i) {
      const int flat = i * 256 + t;
      const int j    = flat >> 2;
      const int sg   = (flat & 3) * 4;
#if QC_ASYNC_LDS
      __builtin_amdgcn_global_load_async_to_lds_b128(
          (gv4i*)(UrT + (size_t)j * DIM + k0 + sg),
          (lv4i*)&UrC[buf][j][sg], 0, 0);
      __builtin_amdgcn_global_load_async_to_lds_b128(
          (gv4i*)(UiT + (size_t)j * DIM + k0 + sg),
          (lv4i*)&UiC[buf][j][sg], 0, 0);
#else
      *(float4*)&UrC[buf][j][sg] = *(const float4*)(UrT + (size_t)j * DIM + k0 + sg);
      *(float4*)&UiC[buf][j][sg] = *(const float4*)(UiT + (size_t)j * DIM + k0 + sg);
#endif
    }
  };
  auto wait_async = [&]() {
#if QC_ASYNC_LDS
    __builtin_amdgcn_s_wait_asynccnt(0);
#endif
  };

  stage(0, 0);
  wait_async();
  __syncthreads();   // chunk 0 staged, A_lds ready

  // ---- Phase 2: WMMA GEMM over K-chunks, Re & Im, both j-tiles live ----
  // f32 16x16x4 fragment layout (ISA 7.12.2): lane m=lane&15 carries row M=m
  // (A) / col N=m (B); lane-hi selects K={0,1} vs K={2,3} across 2 VGPRs.
  const int lane = t & 31;
  const int wv   = t >> 5;
  const int m    = lane & 15;
  const int hi   = lane >> 4;
  const int j0a  = (wv * 2 + 0) * 16;
  const int j0b  = (wv * 2 + 1) * 16;

  v8f aR0 = {}, aI0 = {}, aR1 = {}, aI1 = {};

  for (int c = 0; c < NCHUNK; ++c) {
    const int buf = c & 1;
    if (c + 1 < NCHUNK) stage(buf ^ 1, (c + 1) * KC);   // prefetch next chunk

    const float* rowAr = &UrC[buf][j0a + m][2 * hi];
    const float* rowAi = &UiC[buf][j0a + m][2 * hi];
    const float* rowBr = &UrC[buf][j0b + m][2 * hi];
    const float* rowBi = &UiC[buf][j0b + m][2 * hi];
    const float* arow  = &A_lds[m][c * KC + 2 * hi];
#pragma unroll
    for (int ks = 0; ks < KC; ks += 4) {
      const v2f a   = *reinterpret_cast<const v2f*>(arow + ks);   // ds_load_b64
      const v2f b0r = *reinterpret_cast<const v2f*>(rowAr + ks);
      const v2f b0i = *reinterpret_cast<const v2f*>(rowAi + ks);
      const v2f b1r = *reinterpret_cast<const v2f*>(rowBr + ks);
      const v2f b1i = *reinterpret_cast<const v2f*>(rowBi + ks);
      aR0 = __builtin_amdgcn_wmma_f32_16x16x4_f32(false, a, false, b0r, (short)0, aR0, false, false);
      aI0 = __builtin_amdgcn_wmma_f32_16x16x4_f32(false, a, false, b0i, (short)0, aI0, false, false);
      aR1 = __builtin_amdgcn_wmma_f32_16x16x4_f32(false, a, false, b1r, (short)0, aR1, false, false);
      aI1 = __builtin_amdgcn_wmma_f32_16x16x4_f32(false, a, false, b1i, (short)0, aI1, false, false);
    }

    wait_async();      // next chunk landed in LDS (this wave's share)
    __syncthreads();   // all waves done reading buf & all shares visible
  }

  // C/D layout: VGPR r, lane<16 -> (M=r, N=lane); lane>=16 -> (M=r+8, N=lane-16)
#pragma unroll
  for (int r = 0; r < 8; ++r) {
    const float p0r = aR0[r], p0i = aI0[r];
    const float p1r = aR1[r], p1i = aI1[r];
    P_lds[r + 8 * hi][j0a + m] = p0r * p0r + p0i * p0i;
    P_lds[r + 8 * hi][j0b + m] = p1r * p1r + p1i * p1i;
  }
  __syncthreads();

  // ---- Phase 3: <Z_q> = sum_j sign * p[m][j] ----
  if (t < 128) {
    const int sm = t >> 3, q = t & 7;
    const int sh = 7 - q;
    float acc = 0.f;
#pragma unroll 8
    for (int j = 0; j < DIM; ++j) {
      const float v = P_lds[sm][j];
      acc += ((j >> sh) & 1) ? -v : v;
    }
    out[(size_t)(b0 + sm) * 8 + q] = acc;
  }
}

// ============================== Launcher ===================================
extern "C" void kernel_launch(void* const* d_in, const int* in_sizes, int n_in,
                              void* d_out, int out_size, void* d_ws, size_t ws_size,
                              hipStream_t stream) {
  const float* x     = (const float*)d_in[0];   // [16384, 8] f32
  const float* theta = (const float*)d_in[1];   // [2, 8, 3]  f32
  float* UrT = (float*)d_ws;                    // 256*256 f32
  float* UiT = UrT + DIM * DIM;                 // 256*256 f32 (512 KB total)

  qc_build_unitary<<<DIM, 128, 0, stream>>>(theta, UrT, UiT);
  qc_batch_apply<<<BATCH / 16, 256, 0, stream>>>(x, UrT, UiT, (float*)d_out);
}